// EMDLoss_89593017795422
// MI455X (gfx1250) — compile-verified
//
#include <hip/hip_runtime.h>
#include <math.h>

// ---------------------------------------------------------------------------
// Sinkhorn EMD, kernel-matrix form, LDS-resident, WMMA matvecs (gfx1250).
//   B = 2048 blocks, N = 150 particles padded to NP = 160 (10 tiles of 16).
//   K = exp(-C/eps) built once in LDS; 100 iterations of
//       u = a * rcp(K v);  v = b * rcp(K^T u)
//   as V_WMMA_F32_16X16X4_F32 matvecs, one wave per 16-row tile, with the
//   scaling update fused into the tile-owner wave (2 barriers/iteration).
// ---------------------------------------------------------------------------

typedef float v2f __attribute__((ext_vector_type(2)));
typedef float v8f __attribute__((ext_vector_type(8)));

#define NPART   150
#define NP      160           // padded particle count
#define NTILES  10            // NP / 16
#define KCH     40            // NP / 4 (WMMA k-chunks)
#define KSTR    164           // K row stride in floats (bank-conflict padding)
#define THREADS 320           // 10 waves of 32
#define SINK_ITERS 100
#define SEPS    0.05f
#define EPSF    1e-16f
#define PI_F    3.14159265358979323846f

__device__ __forceinline__ float warp_sum(float x) {
    #pragma unroll
    for (int o = 16; o > 0; o >>= 1) x += __shfl_xor(x, o, 32);
    return x;
}

__global__ void emd_zero_kernel(float* out, int n) {
    int i = blockIdx.x * blockDim.x + threadIdx.x;
    if (i < n) out[i] = 0.0f;
}

__global__ void __launch_bounds__(THREADS, 1)
emd_sinkhorn_kernel(const float* __restrict__ p_recons,
                    const float* __restrict__ p_target,
                    float* __restrict__ out)
{
    extern __shared__ float smem[];
    float* Ksh = smem;                    // NP*KSTR  kernel matrix
    float* XE  = Ksh + NP * KSTR;         // raw px(X) -> eta_rel(X)
    float* XP  = XE + NP;                 // raw py(X) -> phi_rel(X)
    float* XT  = XP + NP;                 // raw pz(X) -> pt_rel(X)
    float* YE  = XT + NP;                 // raw px(Y) -> eta_rel(Y)
    float* YP  = YE + NP;                 // raw py(Y) -> phi_rel(Y)
    float* YT  = YP + NP;                 // raw pz(Y) -> pt_rel(Y)
    float* Aw  = YT + NP;                 // marginal a
    float* Bw  = Aw + NP;                 // marginal b
    float* Uv  = Bw + NP;                 // u scaling
    float* Vv  = Uv + NP;                 // v scaling
    float* red = Vv + NP;                 // 32 floats reduction scratch

    const int tid  = threadIdx.x;
    const int lane = tid & 31;
    const int wave = tid >> 5;
    const int b    = blockIdx.x;

    const float* prb = p_recons + (size_t)b * NPART * 3;
    const float* ptb = p_target + (size_t)b * NPART * 3;

    if (tid == 0) {                       // global_prefetch_b8
        __builtin_prefetch(prb, 0, 0);
        __builtin_prefetch(ptb, 0, 0);
    }

    // ---- stage raw momenta into LDS (pad rows zeroed) ----------------------
    if (tid < NPART) {
        XE[tid] = prb[3 * tid + 0];
        XP[tid] = prb[3 * tid + 1];
        XT[tid] = prb[3 * tid + 2];
        YE[tid] = ptb[3 * tid + 0];
        YP[tid] = ptb[3 * tid + 1];
        YT[tid] = ptb[3 * tid + 2];
    } else if (tid < NP) {
        XE[tid] = 0.f; XP[tid] = 0.f; XT[tid] = 0.f;
        YE[tid] = 0.f; YP[tid] = 0.f; YT[tid] = 0.f;
        Aw[tid] = 0.f; Bw[tid] = 0.f; Uv[tid] = 0.f; Vv[tid] = 0.f;
    }
    __syncthreads();

    // ---- jet component sums (6 arrays are contiguous: XE + w*NP) -----------
    if (wave < 6) {
        const float* arr = XE + wave * NP;
        float s = 0.f;
        #pragma unroll
        for (int i = lane; i < NP; i += 32) s += arr[i];
        s = warp_sum(s);
        if (lane == 0) red[wave] = s;
    }
    __syncthreads();

    const float jxpx = red[0], jxpy = red[1], jxpz = red[2];
    const float jypx = red[3], jypy = red[4], jypz = red[5];

    // ---- cartesian -> (eta_rel, phi_rel, pt_rel), in place -----------------
    if (tid < NPART) {
        {   // X cloud
            float px = XE[tid], py = XP[tid], pz = XT[tid];
            float pt  = sqrtf(px * px + py * py + EPSF);
            float phi = atan2f(py + EPSF, px + EPSF);
            float eta = asinhf(pz / (pt + EPSF));
            float jpt  = sqrtf(jxpx * jxpx + jxpy * jxpy + EPSF);
            float jphi = atan2f(jxpy + EPSF, jxpx + EPSF);
            float jeta = asinhf(jxpz / (jpt + EPSF));
            float d = fmodf(phi - jphi + PI_F, 2.0f * PI_F);
            if (d < 0.f) d += 2.0f * PI_F;
            XE[tid] = eta - jeta;
            XP[tid] = d - PI_F;
            XT[tid] = pt / (jpt + EPSF);
        }
        {   // Y cloud
            float px = YE[tid], py = YP[tid], pz = YT[tid];
            float pt  = sqrtf(px * px + py * py + EPSF);
            float phi = atan2f(py + EPSF, px + EPSF);
            float eta = asinhf(pz / (pt + EPSF));
            float jpt  = sqrtf(jypx * jypx + jypy * jypy + EPSF);
            float jphi = atan2f(jypy + EPSF, jypx + EPSF);
            float jeta = asinhf(jypz / (jpt + EPSF));
            float d = fmodf(phi - jphi + PI_F, 2.0f * PI_F);
            if (d < 0.f) d += 2.0f * PI_F;
            YE[tid] = eta - jeta;
            YP[tid] = d - PI_F;
            YT[tid] = pt / (jpt + EPSF);
        }
    }
    __syncthreads();

    // ---- pt_rel sums -> normalized marginals a, b; init v = 1 --------------
    if (wave < 2) {
        const float* arr = (wave == 0) ? XT : YT;
        float s = 0.f;
        #pragma unroll
        for (int i = lane; i < NP; i += 32) s += arr[i];
        s = warp_sum(s);
        if (lane == 0) red[6 + wave] = s;
    }
    __syncthreads();
    if (tid < NPART) {
        Aw[tid] = XT[tid] * __builtin_amdgcn_rcpf(red[6] + EPSF);
        Bw[tid] = YT[tid] * __builtin_amdgcn_rcpf(red[7] + EPSF);
        Vv[tid] = 1.0f;                    // v0 = exp(g0/eps) = 1
    }
    __syncthreads();

    // ---- build K = exp(-C/eps) once (52M exp2 total vs 9.2G log-domain) ----
    const float kscale = -1.4426950408889634f / SEPS;   // -log2(e)/eps
    for (int idx = tid; idx < NP * NP; idx += THREADS) {
        int i = idx / NP;
        int j = idx - i * NP;
        float de = XE[i] - YE[j];
        float dp = XP[i] - YP[j];
        float C  = sqrtf(de * de + dp * dp + EPSF);
        float k  = exp2f(C * kscale);
        if (i >= NPART || j >= NPART) k = 0.f;          // zero padding rows/cols
        Ksh[i * KSTR + j] = k;
    }
    __syncthreads();

    // ---- 100 Sinkhorn iterations: WMMA matvecs, fused scaling updates ------
    // A-tile (16x4 f32): lanes 0-15 hold row m, k={0,1}; lanes 16-31 k={2,3}.
    // D (16x16 f32): with B replicated across columns, every lane's 8 VGPRs
    // hold rows 8*half+0..7 of the matvec result -> tile-owner update, no LDS
    // round-trip, 2 barriers per iteration.
    const int half = lane >> 4;
    const int m16  = lane & 15;
    const int row0 = wave * 16;            // wave < 10 == NTILES
    const int urow = row0 + 8 * half;

    const float* krow = Ksh + (row0 + m16) * KSTR + 2 * half;

    for (int it = 0; it < SINK_ITERS; ++it) {
        // ---- u = a * rcp(K v) ---------------------------------------------
        {
            v8f acc0 = {0.f, 0.f, 0.f, 0.f, 0.f, 0.f, 0.f, 0.f};
            v8f acc1 = {0.f, 0.f, 0.f, 0.f, 0.f, 0.f, 0.f, 0.f};
            #pragma unroll 4
            for (int kk = 0; kk < KCH; kk += 2) {
                v2f A0; A0.x = krow[4 * kk];     A0.y = krow[4 * kk + 1];
                v2f B0; B0.x = Vv[4 * kk + 2 * half];
                        B0.y = Vv[4 * kk + 2 * half + 1];
                acc0 = __builtin_amdgcn_wmma_f32_16x16x4_f32(
                        false, A0, false, B0, (short)0, acc0, false, false);
                v2f A1; A1.x = krow[4 * kk + 4]; A1.y = krow[4 * kk + 5];
                v2f B1; B1.x = Vv[4 * kk + 4 + 2 * half];
                        B1.y = Vv[4 * kk + 4 + 2 * half + 1];
                acc1 = __builtin_amdgcn_wmma_f32_16x16x4_f32(
                        false, A1, false, B1, (short)0, acc1, false, false);
            }
            if (m16 == 0) {                // lanes 0 & 16 own rows urow..urow+7
                #pragma unroll
                for (int r = 0; r < 8; ++r) {
                    float rv = acc0[r] + acc1[r];
                    Uv[urow + r] = Aw[urow + r] *
                                   __builtin_amdgcn_rcpf(rv + 1e-30f);
                }
            }
        }
        __syncthreads();

        // ---- v = b * rcp(K^T u) -------------------------------------------
        {
            v8f acc0 = {0.f, 0.f, 0.f, 0.f, 0.f, 0.f, 0.f, 0.f};
            v8f acc1 = {0.f, 0.f, 0.f, 0.f, 0.f, 0.f, 0.f, 0.f};
            #pragma unroll 4
            for (int kk = 0; kk < KCH; kk += 2) {
                const float* kc0 = Ksh + (4 * kk + 2 * half) * KSTR + row0 + m16;
                v2f A0; A0.x = kc0[0];    A0.y = kc0[KSTR];
                v2f B0; B0.x = Uv[4 * kk + 2 * half];
                        B0.y = Uv[4 * kk + 2 * half + 1];
                acc0 = __builtin_amdgcn_wmma_f32_16x16x4_f32(
                        false, A0, false, B0, (short)0, acc0, false, false);
                const float* kc1 = kc0 + 4 * KSTR;
                v2f A1; A1.x = kc1[0];    A1.y = kc1[KSTR];
                v2f B1; B1.x = Uv[4 * kk + 4 + 2 * half];
                        B1.y = Uv[4 * kk + 4 + 2 * half + 1];
                acc1 = __builtin_amdgcn_wmma_f32_16x16x4_f32(
                        false, A1, false, B1, (short)0, acc1, false, false);
            }
            if (m16 == 0) {
                #pragma unroll
                for (int r = 0; r < 8; ++r) {
                    float sv = acc0[r] + acc1[r];
                    Vv[urow + r] = Bw[urow + r] *
                                   __builtin_amdgcn_rcpf(sv + 1e-30f);
                }
            }
        }
        __syncthreads();
    }

    // ---- loss = sum_ij u_i K_ij v_j C_ij (C recomputed on the fly) ---------
    float partial = 0.f;
    for (int idx = tid; idx < NP * NP; idx += THREADS) {
        int i = idx / NP;
        int j = idx - i * NP;
        float de = XE[i] - YE[j];
        float dp = XP[i] - YP[j];
        float C  = sqrtf(de * de + dp * dp + EPSF);
        partial += Uv[i] * Ksh[i * KSTR + j] * Vv[j] * C;
    }
    partial = warp_sum(partial);
    if (lane == 0) red[8 + wave] = partial;
    __syncthreads();
    if (wave == 0) {
        float s = (lane < NTILES) ? red[8 + lane] : 0.f;
        s = warp_sum(s);
        if (lane == 0) atomicAdd(out, s);
    }
}

extern "C" void kernel_launch(void* const* d_in, const int* in_sizes, int n_in,
                              void* d_out, int out_size, void* d_ws, size_t ws_size,
                              hipStream_t stream) {
    (void)n_in; (void)d_ws; (void)ws_size;
    const float* p_recons = (const float*)d_in[0];
    const float* p_target = (const float*)d_in[1];
    float* out = (float*)d_out;

    const int B = in_sizes[0] / (NPART * 3);   // 2048

    emd_zero_kernel<<<1, THREADS, 0, stream>>>(out, out_size);

    const size_t smem_bytes = (size_t)(NP * KSTR + 10 * NP + 32) * sizeof(float);
    emd_sinkhorn_kernel<<<B, THREADS, smem_bytes, stream>>>(p_recons, p_target, out);
}